// QuPoolXYZ_46952582479902
// MI455X (gfx1250) — compile-verified
//
#include <hip/hip_runtime.h>

typedef __attribute__((ext_vector_type(2))) float v2f;
typedef __attribute__((ext_vector_type(8))) float v8f;

#define NQ    10
#define DIM   1024
#define NP    57
#define W_MUL 0.6324555320336759f   // sqrt(2/5)

// One workgroup per batch element. 256 threads = 8 wave32s.
// Phase 1: state-vector evolution in LDS (9 fused pair gates + 10 fused 1q gates).
// Phase 2: 32x32 Gram matrix red[a,c] = sum_i psi[i*32+a]*conj(psi[i*32+c])
//          via V_WMMA_F32_16X16X4_F32 (4 tiles x {re,im} across the 8 waves).
__global__ __launch_bounds__(256) void qupool_kernel(
    const float* __restrict__ xr, const float* __restrict__ xi,
    const float* __restrict__ w, float* __restrict__ out)
{
    __shared__ float sre[DIM];
    __shared__ float sim[DIM];
    __shared__ float cH[NP];   // cos(theta/2)
    __shared__ float sH[NP];   // sin(theta/2)

    const int t = threadIdx.x;
    const int b = blockIdx.x;

    // ---- precompute half-angle cos/sin once per block ----
    if (t < NP) {
        float th = w[t] * (0.5f * W_MUL);
        float s, c;
        sincosf(th, &s, &c);
        cH[t] = c; sH[t] = s;
    }
    // ---- load |x> into LDS (psi) ----
    for (int j = t; j < DIM; j += 256) {
        sre[j] = xr[b * DIM + j];
        sim[j] = xi[b * DIM + j];
    }
    __syncthreads();

    // ---- 9 fused two-qubit gates: U = CNOT * Rzz * Ryy * Rxx on bits (9-q, 8-q)
    #pragma unroll
    for (int q = 0; q < NQ - 1; ++q) {
        const float cx = cH[3*q+0], sx = sH[3*q+0];
        const float cy = cH[3*q+1], sy = sH[3*q+1];
        const float cz = cH[3*q+2], sz = sH[3*q+2];
        // {|00>,|11>} block: (cz - i sz) * [[P, iQ],[iQ, P]]
        // {|01>,|10>} block: (cz + i sz) * [[R,-iS],[-iS, R]]
        const float P = cx*cy + sx*sy, Q = sy*cx - sx*cy;
        const float R = cx*cy - sx*sy, S = sx*cy + sy*cx;

        const int bl  = 8 - q;                       // lower bit of the pair
        const int g   = t;                           // 256 disjoint groups of 4 amps
        const int low = g & ((1 << bl) - 1);
        const int i00 = ((g >> bl) << (bl + 2)) | low;
        const int i01 = i00 | (1 << bl);
        const int i10 = i00 | (2 << bl);
        const int i11 = i00 | (3 << bl);

        const float v00r = sre[i00], v00i = sim[i00];
        const float v01r = sre[i01], v01i = sim[i01];
        const float v10r = sre[i10], v10i = sim[i10];
        const float v11r = sre[i11], v11i = sim[i11];

        // u00 = (cz - i sz)*(P*v00 + iQ*v11)
        float t0r = P*v00r - Q*v11i, t0i = P*v00i + Q*v11r;
        float u00r = cz*t0r + sz*t0i, u00i = cz*t0i - sz*t0r;
        // u11 = (cz - i sz)*(iQ*v00 + P*v11)
        float t1r = P*v11r - Q*v00i, t1i = P*v11i + Q*v00r;
        float u11r = cz*t1r + sz*t1i, u11i = cz*t1i - sz*t1r;
        // u01 = (cz + i sz)*(R*v01 - iS*v10)
        float t2r = R*v01r + S*v10i, t2i = R*v01i - S*v10r;
        float u01r = cz*t2r - sz*t2i, u01i = cz*t2i + sz*t2r;
        // u10 = (cz + i sz)*(-iS*v01 + R*v10)
        float t3r = R*v10r + S*v01i, t3i = R*v10i - S*v01r;
        float u10r = cz*t3r - sz*t3i, u10i = cz*t3i + sz*t3r;

        sre[i00] = u00r; sim[i00] = u00i;
        sre[i01] = u01r; sim[i01] = u01i;
        sre[i10] = u11r; sim[i10] = u11i;   // CNOT swaps |10> <-> |11>
        sre[i11] = u10r; sim[i11] = u10i;
        __syncthreads();
    }

    // ---- 10 fused single-qubit gates: U = Rx * Ry * Rz on bit (9-q)
    #pragma unroll
    for (int q = 0; q < NQ; ++q) {
        const float cz = cH[27+3*q+0], sz = sH[27+3*q+0];
        const float cy = cH[27+3*q+1], sy = sH[27+3*q+1];
        const float cx = cH[27+3*q+2], sx = sH[27+3*q+2];
        // Ry*Rz
        const float m00r =  cy*cz, m00i = -cy*sz;
        const float m01r = -sy*cz, m01i = -sy*sz;
        const float m10r =  sy*cz, m10i = -sy*sz;
        const float m11r =  cy*cz, m11i =  cy*sz;
        // Rx * (Ry*Rz)
        const float U00r = cx*m00r + sx*m10i, U00i = cx*m00i - sx*m10r;
        const float U01r = cx*m01r + sx*m11i, U01i = cx*m01i - sx*m11r;
        const float U10r = sx*m00i + cx*m10r, U10i = -sx*m00r + cx*m10i;
        const float U11r = sx*m01i + cx*m11r, U11i = -sx*m01r + cx*m11i;

        const int k = 9 - q;
        #pragma unroll
        for (int rep = 0; rep < 2; ++rep) {
            const int p   = t + rep * 256;           // 512 disjoint pairs
            const int low = p & ((1 << k) - 1);
            const int i0  = ((p >> k) << (k + 1)) | low;
            const int i1  = i0 | (1 << k);
            const float ar = sre[i0], ai = sim[i0];
            const float br = sre[i1], bi = sim[i1];
            sre[i0] = U00r*ar - U00i*ai + U01r*br - U01i*bi;
            sim[i0] = U00r*ai + U00i*ar + U01r*bi + U01i*br;
            sre[i1] = U10r*ar - U10i*ai + U11r*br - U11i*bi;
            sim[i1] = U10r*ai + U10i*ar + U11r*bi + U11i*br;
        }
        __syncthreads();
    }

    // ---- Gram matrix via f32 WMMA: red[a,c] = sum_i psi[i*32+a]*conj(psi[i*32+c])
    // 8 waves: tile = wave&3 -> 16x16 output tile (ta,tc); part = wave>>2 -> re/im.
    // re = Pr^T Pr + Pi^T Pi ; im = Pi^T Pr + (-Pr)^T Pi
    const int lane = t & 31;
    const int wave = t >> 5;
    const int ta   = (wave & 3) >> 1;
    const int tc   = (wave & 3) & 1;
    const int part = wave >> 2;               // 0 = real, 1 = imag
    const int col  = lane & 15;
    const int kb   = (lane >> 4) << 1;        // A/B per-lane K base: 0 or 2
    const int aCol = 16 * ta + col;           // A[m,k] = psi[k*32 + 16*ta + m]
    const int bCol = 16 * tc + col;           // B[k,n] = psi[k*32 + 16*tc + n]

    const float* Asrc1 = part ? sim : sre;    // pass 1: A
    const float* Bsrc1 = sre;                 // pass 1: B
    const float* Asrc2 = part ? sre : sim;    // pass 2: A (negated for imag)
    const float* Bsrc2 = sim;                 // pass 2: B
    const float  aSgn2 = part ? -1.0f : 1.0f;

    v8f acc = {};
    #pragma unroll
    for (int k0 = 0; k0 < 32; k0 += 4) {
        v2f A, B;
        A.x = Asrc1[(k0 + kb)     * 32 + aCol];
        A.y = Asrc1[(k0 + kb + 1) * 32 + aCol];
        B.x = Bsrc1[(k0 + kb)     * 32 + bCol];
        B.y = Bsrc1[(k0 + kb + 1) * 32 + bCol];
        acc = __builtin_amdgcn_wmma_f32_16x16x4_f32(false, A, false, B,
                                                    (short)0, acc, false, false);
    }
    #pragma unroll
    for (int k0 = 0; k0 < 32; k0 += 4) {
        v2f A, B;
        A.x = aSgn2 * Asrc2[(k0 + kb)     * 32 + aCol];
        A.y = aSgn2 * Asrc2[(k0 + kb + 1) * 32 + aCol];
        B.x = Bsrc2[(k0 + kb)     * 32 + bCol];
        B.y = Bsrc2[(k0 + kb + 1) * 32 + bCol];
        acc = __builtin_amdgcn_wmma_f32_16x16x4_f32(false, A, false, B,
                                                    (short)0, acc, false, false);
    }

    // D layout: lanes 0-15 -> (M=j,   N=lane); lanes 16-31 -> (M=j+8, N=lane-16)
    #pragma unroll
    for (int j = 0; j < 8; ++j) {
        const int M = (lane < 16) ? j : (j + 8);
        const int a = 16 * ta + M;
        const int c = 16 * tc + (lane & 15);
        out[(size_t)(b * 1024 + a * 32 + c) * 2 + part] = acc[j];
    }
}

extern "C" void kernel_launch(void* const* d_in, const int* in_sizes, int n_in,
                              void* d_out, int out_size, void* d_ws, size_t ws_size,
                              hipStream_t stream) {
    const float* xr = (const float*)d_in[0];   // (B,1,1024) f32
    const float* xi = (const float*)d_in[1];   // (B,1,1024) f32
    const float* w  = (const float*)d_in[2];   // (57,) f32
    float* out = (float*)d_out;                // (B,1,1024) complex64 -> interleaved f32
    const int B = in_sizes[0] / DIM;           // 16
    qupool_kernel<<<B, 256, 0, stream>>>(xr, xi, w, out);
}